// SimpleSequenceSummarizationBlock_11879879541476
// MI455X (gfx1250) — compile-verified
//
#include <hip/hip_runtime.h>
#include <cstdint>

typedef __attribute__((ext_vector_type(2))) float v2f;
typedef __attribute__((ext_vector_type(8))) float v8f;

#define B_    512
#define S_    1000
#define D_    384
#define K_SEL 120
#define LN_EPS 1e-5f

// ---------------------------------------------------------------------------
// Kernel 1: routing_w[b,s] = x[b,s,:].router_w + router_b ; zero coeff[b,s]
// One wave (32 lanes) per token; float4 coalesced loads; shfl reduction.
// ---------------------------------------------------------------------------
__global__ __launch_bounds__(256) void router_kernel(
    const float* __restrict__ x,
    const float* __restrict__ rw,
    const float* __restrict__ rb,
    float* __restrict__ routing,
    float4* __restrict__ coeff) {
  int wv   = threadIdx.x >> 5;
  int lane = threadIdx.x & 31;
  long t = (long)blockIdx.x * 8 + wv;          // token id
  if (t >= (long)B_ * S_) return;
  const float4* xr  = (const float4*)(x + t * D_);
  const float4* wr4 = (const float4*)rw;
  float acc = 0.f;
  #pragma unroll
  for (int i = 0; i < 3; ++i) {                // 384/4 = 96 float4, 3 per lane
    int idx = lane + 32 * i;
    float4 a = xr[idx];
    float4 b = wr4[idx];
    acc += a.x * b.x + a.y * b.y + a.z * b.z + a.w * b.w;
  }
  #pragma unroll
  for (int m = 16; m >= 1; m >>= 1) acc += __shfl_xor(acc, m, 32);
  if (lane == 0) {
    routing[t] = acc + rb[0];
    coeff[t]   = make_float4(0.f, 0.f, 0.f, 0.f);
  }
}

// ---------------------------------------------------------------------------
// Kernel 2: per-batch-row top-k(120), sort indices ascending, compute LN
// affine coefficients for selected tokens. One 512-thread block per b.
// ---------------------------------------------------------------------------
__global__ __launch_bounds__(512) void topk_kernel(
    const float* __restrict__ x,
    const float* __restrict__ routing,
    float4* __restrict__ coeff) {
  __shared__ unsigned long long keys[1024];
  __shared__ unsigned long long sel[128];
  int tid = threadIdx.x;
  int b   = blockIdx.x;
  const float* row = routing + (long)b * S_;

  // pack: key = (orderable(value) << 32) | ~index   (tie-break: lower index wins)
  for (int i = tid; i < 1024; i += 512) {
    unsigned long long key = 0ULL;               // padding sorts last (descending)
    if (i < S_) {
      unsigned u = __float_as_uint(row[i]);
      u = (u & 0x80000000u) ? ~u : (u | 0x80000000u);
      key = ((unsigned long long)u << 32) | (unsigned)(~(unsigned)i);
    }
    keys[i] = key;
  }
  __syncthreads();

  // bitonic sort 1024, DESCENDING
  for (int k = 2; k <= 1024; k <<= 1) {
    for (int j = k >> 1; j > 0; j >>= 1) {
      #pragma unroll
      for (int base = 0; base < 1024; base += 512) {
        int i = base + tid;
        int l = i ^ j;
        if (l > i) {
          bool up = ((i & k) == 0);
          unsigned long long a = keys[i], c = keys[l];
          bool sw = up ? (a < c) : (a > c);      // descending network
          if (sw) { keys[i] = c; keys[l] = a; }
        }
      }
      __syncthreads();
    }
  }

  // take top-120, repack as (index << 32) | weight_bits for ascending index sort
  if (tid < 128) {
    unsigned long long k2 = 0xFFFFFFFFFFFFFFFFULL;   // padding sorts last (ascending)
    if (tid < K_SEL) {
      unsigned long long kk = keys[tid];
      unsigned u    = (unsigned)(kk >> 32);
      unsigned bits = (u & 0x80000000u) ? (u ^ 0x80000000u) : ~u;
      unsigned sidx = ~(unsigned)(kk & 0xFFFFFFFFu);
      k2 = ((unsigned long long)sidx << 32) | (unsigned long long)bits;
    }
    sel[tid] = k2;
  }
  __syncthreads();

  // bitonic sort 128, ASCENDING (indices unique)
  for (int k = 2; k <= 128; k <<= 1) {
    for (int j = k >> 1; j > 0; j >>= 1) {
      if (tid < 128) {
        int i = tid, l = i ^ j;
        if (l > i && l < 128) {
          bool up = ((i & k) == 0);
          unsigned long long a = sel[i], c = sel[l];
          bool sw = up ? (a > c) : (a < c);
          if (sw) { sel[i] = c; sel[l] = a; }
        }
      }
      __syncthreads();
    }
  }

  // per selected token: mean/var over D, write affine coeffs
  int wv = tid >> 5, ln = tid & 31;               // 16 waves
  for (int j = wv; j < K_SEL; j += 16) {
    unsigned long long k2 = sel[j];
    int sidx = (int)(k2 >> 32);
    float w  = __uint_as_float((unsigned)(k2 & 0xFFFFFFFFu));
    const float* xr = x + ((long)b * S_ + sidx) * D_;
    float s = 0.f, ss = 0.f;
    #pragma unroll
    for (int u2 = 0; u2 < 12; ++u2) {
      float v = xr[ln + 32 * u2];
      s += v; ss += v * v;
    }
    #pragma unroll
    for (int m = 16; m >= 1; m >>= 1) {
      s  += __shfl_xor(s,  m, 32);
      ss += __shfl_xor(ss, m, 32);
    }
    if (ln == 0) {
      float mu  = s / (float)D_;
      float var = ss / (float)D_ - mu * mu;
      float r   = rsqrtf(var + LN_EPS);
      // x_full[d] = x[d]*(1 + c1*gamma[d]) + (c2*gamma[d] + c3*beta[d])
      coeff[(long)b * S_ + sidx] = make_float4(w * r, -w * r * mu, w, 0.f);
    }
  }
}

// ---------------------------------------------------------------------------
// Kernel 3: out[m,n] = sum_k x_eff[m,k] * gate_w[n,k] + gate_b[n]
// M=512000, N=384, K=384. Block tile 128x64, K staged 32 via LDS.
// 8 waves (4M x 2N), each wave a 32x32 tile = 2x2 fragments of 16x16,
// v_wmma_f32_16x16x4_f32 accumulation. LN update fused into A-tile load.
// ---------------------------------------------------------------------------
#define MB  128
#define NB  64
#define KB  32
#define LDA 36
#define LDB 36

__global__ __launch_bounds__(256) void gemm_kernel(
    const float* __restrict__ x,
    const float* __restrict__ gate_w,
    const float* __restrict__ gate_b,
    const float* __restrict__ gamma,
    const float* __restrict__ beta,
    const float4* __restrict__ coeff,
    float* __restrict__ out) {
  __shared__ float  As[MB * LDA];     // 18 KB
  __shared__ float  Bs[NB * LDB];     //  9 KB
  __shared__ float4 cf[MB];           //  2 KB
  __shared__ float  gLds[D_];         // 1.5 KB
  __shared__ float  bLds[D_];         // 1.5 KB

  int tid = threadIdx.x;
  long m0 = (long)blockIdx.x * MB;
  int  n0 = blockIdx.y * NB;

  for (int i = tid; i < D_; i += 256) { gLds[i] = gamma[i]; bLds[i] = beta[i]; }
  if (tid < MB) cf[tid] = coeff[m0 + tid];

  int lane = tid & 31;
  int wave = tid >> 5;
  int wm   = (wave >> 1) * 32;        // 0,32,64,96
  int wn   = (wave & 1) * 32;         // 0,32
  int hlf  = lane >> 4;               // K-halves of the A/B fragments
  int l16  = lane & 15;

  v8f acc[2][2];
  #pragma unroll
  for (int im = 0; im < 2; ++im)
    #pragma unroll
    for (int in = 0; in < 2; ++in)
      acc[im][in] = (v8f){0.f, 0.f, 0.f, 0.f, 0.f, 0.f, 0.f, 0.f};

  int tx = tid & 7;                   // k col-group (4 floats)
  int ty = tid >> 3;                  // 0..31
  int colA = tx * 4;

  __syncthreads();                    // cf / gamma / beta visible

  const int KSTAGES = D_ / KB;        // 12
  for (int st = 0; st < KSTAGES; ++st) {
    int kb = st * KB;

    // --- A tile load: apply fused LN affine update per element ---
    float4 g4 = *(const float4*)&gLds[kb + colA];
    float4 b4 = *(const float4*)&bLds[kb + colA];
    #pragma unroll
    for (int i = 0; i < 4; ++i) {
      int r = ty + 32 * i;
      float4 c = cf[r];
      float4 v = *(const float4*)(x + (m0 + r) * D_ + kb + colA);
      float4 o;
      o.x = v.x * (1.f + c.x * g4.x) + (c.y * g4.x + c.z * b4.x);
      o.y = v.y * (1.f + c.x * g4.y) + (c.y * g4.y + c.z * b4.y);
      o.z = v.z * (1.f + c.x * g4.z) + (c.y * g4.z + c.z * b4.z);
      o.w = v.w * (1.f + c.x * g4.w) + (c.y * g4.w + c.z * b4.w);
      *(float4*)&As[r * LDA + colA] = o;
    }
    // --- B tile load: Bs[n][k] = gate_w[n0+n][kb+k] (natural layout) ---
    #pragma unroll
    for (int i = 0; i < 2; ++i) {
      int n = ty + 32 * i;
      *(float4*)&Bs[n * LDB + colA] =
          *(const float4*)(gate_w + (long)(n0 + n) * D_ + kb + colA);
    }
    // prefetch next stage (gfx1250 global_prefetch_b8)
    if (st + 1 < KSTAGES) {
      __builtin_prefetch(x + (m0 + ty) * D_ + kb + KB + colA);
      __builtin_prefetch(gate_w + (long)(n0 + ty) * D_ + kb + KB + colA);
    }
    __syncthreads();

    // --- compute: 8 k-steps of 4, 4 WMMAs each ---
    #pragma unroll
    for (int kk = 0; kk < KB; kk += 4) {
      int ka = kk + hlf * 2;          // lanes 0-15 -> K{0,1}, lanes 16-31 -> K{2,3}
      v2f a[2], bf[2];
      #pragma unroll
      for (int im = 0; im < 2; ++im) {
        int r = wm + im * 16 + l16;   // A: M striped over lanes 0-15
        a[im].x = As[r * LDA + ka];
        a[im].y = As[r * LDA + ka + 1];
      }
      #pragma unroll
      for (int in = 0; in < 2; ++in) {
        int cb = wn + in * 16 + l16;  // B: N striped over lanes, B[k][n]=Bs[n][k]
        bf[in].x = Bs[cb * LDB + ka];
        bf[in].y = Bs[cb * LDB + ka + 1];
      }
      #pragma unroll
      for (int im = 0; im < 2; ++im)
        #pragma unroll
        for (int in = 0; in < 2; ++in)
          acc[im][in] = __builtin_amdgcn_wmma_f32_16x16x4_f32(
              false, a[im], false, bf[in], (short)0, acc[im][in], false, false);
    }
    __syncthreads();
  }

  // --- epilogue: + gate_b, store (C/D layout: vgpr=M row, lanes 16-31 -> M+8) ---
  #pragma unroll
  for (int in = 0; in < 2; ++in) {
    int ng = n0 + wn + in * 16 + l16;
    float gb = gate_b[ng];
    #pragma unroll
    for (int im = 0; im < 2; ++im) {
      long mg = m0 + wm + im * 16 + hlf * 8;
      #pragma unroll
      for (int v = 0; v < 8; ++v)
        out[(mg + v) * D_ + ng] = acc[im][in][v] + gb;
    }
  }
}

// ---------------------------------------------------------------------------
extern "C" void kernel_launch(void* const* d_in, const int* in_sizes, int n_in,
                              void* d_out, int out_size, void* d_ws, size_t ws_size,
                              hipStream_t stream) {
  const float* x        = (const float*)d_in[0];
  const float* router_w = (const float*)d_in[1];
  const float* router_b = (const float*)d_in[2];
  const float* ln_gamma = (const float*)d_in[3];
  const float* ln_beta  = (const float*)d_in[4];
  const float* gate_w   = (const float*)d_in[5];
  const float* gate_b   = (const float*)d_in[6];
  float* out = (float*)d_out;

  char* ws = (char*)d_ws;
  float*  routing = (float*)ws;                                  // 512000 f32 (2 MB)
  float4* coeff   = (float4*)(ws + (size_t)B_ * S_ * sizeof(float)); // 512000 f4 (8 MB)

  // K1: router GEMV (+ zero coeffs). 8 tokens per 256-thread block.
  router_kernel<<<(B_ * S_) / 8, 256, 0, stream>>>(x, router_w, router_b,
                                                   routing, coeff);
  // K2: per-row top-k + index sort + LN coefficients.
  topk_kernel<<<B_, 512, 0, stream>>>(x, routing, coeff);
  // K3: fused (scatter-update ∘ gating GEMM). Grid: 4000 M-tiles x 6 N-tiles.
  dim3 grid((B_ * S_) / MB, D_ / NB);
  gemm_kernel<<<grid, 256, 0, stream>>>(x, gate_w, gate_b, ln_gamma, ln_beta,
                                        coeff, out);
}